// CUDALearnableGaussianField_88064009437940
// MI455X (gfx1250) — compile-verified
//
#include <hip/hip_runtime.h>

// CDNA5 / gfx1250: wave32, WMMA 16x16 tiles.
typedef __attribute__((ext_vector_type(2))) float v2f;
typedef __attribute__((ext_vector_type(8))) float v8f;

#define NPTS   32768
#define NG     4096
#define NSEG   8
#define SEGLEN (NG / NSEG)          // 512 gaussians per segment
#define CHUNKS (SEGLEN / 16)        // 32 chunks of 16 gaussians

// -0.5 * log2(e): folded into the coefficients so WMMA output is the exp2 arg.
#define NEG_HALF_LOG2E (-0.72134752044f)

// Raw hardware exp2: single v_exp_f32, no subnormal-range guard. Arguments
// here are <= ~0 and anything below -126 contributes ~0 to a 4096-term sum,
// so flush-to-zero is fine and we skip OCML's cmp/cndmask/add/scale wrapper.
#if __has_builtin(__builtin_amdgcn_exp2f)
#define FAST_EXP2(v) __builtin_amdgcn_exp2f(v)
#else
#define FAST_EXP2(v) __ocml_exp2_f32(v)
#endif

// ---------------------------------------------------------------------------
// Kernel 1: per-Gaussian coefficient precompute (quadratic-form linearization)
// coef layout: SoA, coef[k*NG + n], k = 0..11
//   k0..2 : x0^2,x1^2,x2^2 coeffs  (A00,A11,A22)
//   k3..5 : x0x1,x0x2,x1x2 coeffs  (2A01,2A02,2A12)
//   k6..8 : linear coeffs          (-2*A*mu)
//   k9    : mu^T A mu - 2*ln(w)
//   k10,11: zero padding (K=12 for three 16x16x4 WMMAs)
// All scaled by -0.5*log2(e).
// ---------------------------------------------------------------------------
__global__ __launch_bounds__(256) void prep_coef(
    const float* __restrict__ means, const float* __restrict__ cov_tril,
    const float* __restrict__ weights, float* __restrict__ coef) {
  int n = blockIdx.x * blockDim.x + threadIdx.x;
  if (n >= NG) return;
  const float eps = 1e-6f;
  float t0 = cov_tril[n * 6 + 0], t1 = cov_tril[n * 6 + 1];
  float t2 = cov_tril[n * 6 + 2], t3 = cov_tril[n * 6 + 3];
  float t4 = cov_tril[n * 6 + 4], t5 = cov_tril[n * 6 + 5];
  float La = __expf(t0) + eps;   // L00
  float Lc = __expf(t2) + eps;   // L11
  float Lf = __expf(t5) + eps;   // L22
  // M = L^-1 (lower triangular)
  float M00 = 1.0f / La, M11 = 1.0f / Lc, M22 = 1.0f / Lf;
  float M10 = -t1 * M00 * M11;
  float M21 = -t4 * M11 * M22;
  float M20 = (t1 * t4 - Lc * t3) * M00 * M11 * M22;
  // A = M^T M (symmetric)
  float A00 = M00 * M00 + M10 * M10 + M20 * M20;
  float A01 = M10 * M11 + M20 * M21;
  float A02 = M20 * M22;
  float A11 = M11 * M11 + M21 * M21;
  float A12 = M21 * M22;
  float A22 = M22 * M22;
  float mu0 = means[n * 3 + 0], mu1 = means[n * 3 + 1], mu2 = means[n * 3 + 2];
  float g0 = A00 * mu0 + A01 * mu1 + A02 * mu2;
  float g1 = A01 * mu0 + A11 * mu1 + A12 * mu2;
  float g2 = A02 * mu0 + A12 * mu1 + A22 * mu2;
  float quad = mu0 * g0 + mu1 * g1 + mu2 * g2;
  float cc = quad - 2.0f * __logf(weights[n]);
  const float s = NEG_HALF_LOG2E;
  coef[0 * NG + n] = s * A00;
  coef[1 * NG + n] = s * A11;
  coef[2 * NG + n] = s * A22;
  coef[3 * NG + n] = s * 2.0f * A01;
  coef[4 * NG + n] = s * 2.0f * A02;
  coef[5 * NG + n] = s * 2.0f * A12;
  coef[6 * NG + n] = s * -2.0f * g0;
  coef[7 * NG + n] = s * -2.0f * g1;
  coef[8 * NG + n] = s * -2.0f * g2;
  coef[9 * NG + n] = s * cc;
  coef[10 * NG + n] = 0.0f;
  coef[11 * NG + n] = 0.0f;
}

// ---------------------------------------------------------------------------
// Kernel 2: mahal GEMM via V_WMMA_F32_16X16X4_F32 (K=12 => 3 WMMAs / tile),
// then raw v_exp_f32 + accumulate. One wave = 16 points x one N-segment.
// A layout (16x4 f32): lanes 0-15 hold {K=0,K=1}, lanes 16-31 hold {K=2,K=3}.
// C layout (16x16 f32): VGPR i, half h -> point row i+8h, column = lane&15.
// ---------------------------------------------------------------------------
__global__ __launch_bounds__(256) void gauss_field(
    const float* __restrict__ x, const float* __restrict__ coef,
    float* __restrict__ partial) {
  const int lane   = threadIdx.x & 31;
  const int waveId = blockIdx.x * 8 + (threadIdx.x >> 5);
  const int seg    = waveId & (NSEG - 1);
  const int tile   = waveId >> 3;
  const int pointBase = tile * 16;
  const int m = lane & 15;   // column / point-in-tile index
  const int h = lane >> 4;   // half: selects K pair

  // Loop-invariant A-matrix fragments (features of this lane's point).
  const int p = pointBase + m;
  float x0 = x[p * 3 + 0], x1 = x[p * 3 + 1], x2 = x[p * 3 + 2];
  float f[12];
  f[0] = x0 * x0; f[1] = x1 * x1; f[2] = x2 * x2;
  f[3] = x0 * x1; f[4] = x0 * x2; f[5] = x1 * x2;
  f[6] = x0;      f[7] = x1;      f[8] = x2;
  f[9] = 1.0f;    f[10] = 0.0f;   f[11] = 0.0f;
  v2f a0 = {f[2 * h + 0], f[2 * h + 1]};          // K 0..3
  v2f a1 = {f[4 + 2 * h + 0], f[4 + 2 * h + 1]};  // K 4..7
  v2f a2 = {f[8 + 2 * h + 0], f[8 + 2 * h + 1]};  // K 8..11

  // B-matrix row pointers for this lane's half (coalesced across lanes).
  const float* c0 = coef + (0 + 2 * h) * NG;
  const float* c1 = coef + (1 + 2 * h) * NG;
  const float* c2 = coef + (4 + 2 * h) * NG;
  const float* c3 = coef + (5 + 2 * h) * NG;
  const float* c4 = coef + (8 + 2 * h) * NG;
  const float* c5 = coef + (9 + 2 * h) * NG;

  float acc[8];
#pragma unroll
  for (int i = 0; i < 8; ++i) acc[i] = 0.0f;

  const int nBase = seg * SEGLEN + m;
#pragma unroll 4
  for (int chunk = 0; chunk < CHUNKS; ++chunk) {
    const int n0 = nBase + chunk * 16;
    v2f b0 = {c0[n0], c1[n0]};
    v2f b1 = {c2[n0], c3[n0]};
    v2f b2 = {c4[n0], c5[n0]};
    v8f cm = {};
    cm = __builtin_amdgcn_wmma_f32_16x16x4_f32(false, a0, false, b0,
                                               (short)0, cm, false, false);
    cm = __builtin_amdgcn_wmma_f32_16x16x4_f32(false, a1, false, b1,
                                               (short)0, cm, false, false);
    cm = __builtin_amdgcn_wmma_f32_16x16x4_f32(false, a2, false, b2,
                                               (short)0, cm, false, false);
#pragma unroll
    for (int i = 0; i < 8; ++i)
      acc[i] += FAST_EXP2(cm[i]);  // single v_exp_f32 per element
  }

  // Sum over the 16 columns held by lanes within each half.
#pragma unroll
  for (int i = 0; i < 8; ++i) {
    float v = acc[i];
    v += __shfl_xor(v, 1, 32);
    v += __shfl_xor(v, 2, 32);
    v += __shfl_xor(v, 4, 32);
    v += __shfl_xor(v, 8, 32);
    acc[i] = v;
  }
  if (m == 0) {
#pragma unroll
    for (int i = 0; i < 8; ++i)
      partial[seg * NPTS + pointBase + h * 8 + i] = acc[i];
  }
}

// ---------------------------------------------------------------------------
// Kernel 3: deterministic segment reduction.
// ---------------------------------------------------------------------------
__global__ __launch_bounds__(256) void reduce_partials(
    const float* __restrict__ partial, float* __restrict__ out) {
  int p = blockIdx.x * blockDim.x + threadIdx.x;
  if (p >= NPTS) return;
  float s = 0.0f;
#pragma unroll
  for (int i = 0; i < NSEG; ++i) s += partial[i * NPTS + p];
  out[p] = s;
}

extern "C" void kernel_launch(void* const* d_in, const int* in_sizes, int n_in,
                              void* d_out, int out_size, void* d_ws,
                              size_t ws_size, hipStream_t stream) {
  (void)in_sizes; (void)n_in; (void)out_size; (void)ws_size;
  const float* x        = (const float*)d_in[0];  // (32768,3)
  const float* means    = (const float*)d_in[1];  // (4096,3)
  const float* cov_tril = (const float*)d_in[2];  // (4096,6)
  const float* weights  = (const float*)d_in[3];  // (4096,)
  float* out = (float*)d_out;                     // (32768,)

  float* coef    = (float*)d_ws;                  // 12*4096 floats (192 KB)
  float* partial = coef + 12 * NG;                // 8*32768 floats (1 MB)

  prep_coef<<<(NG + 255) / 256, 256, 0, stream>>>(means, cov_tril, weights,
                                                  coef);
  // (NPTS/16 tiles) * NSEG segments = 16384 waves -> 2048 blocks of 8 waves.
  gauss_field<<<(NPTS / 16) * NSEG / 8, 256, 0, stream>>>(x, coef, partial);
  reduce_partials<<<(NPTS + 255) / 256, 256, 0, stream>>>(partial, out);
}